// PointLoss_69870527971439
// MI455X (gfx1250) — compile-verified
//
#include <hip/hip_runtime.h>

typedef __attribute__((ext_vector_type(2))) float v2f;
typedef __attribute__((ext_vector_type(8))) float v8f;

#define TILE 16
#define NPTS 4096
#define NTILES (NPTS / TILE)      // 256
#define NBATCH 4
#define WAVES_PER_BLOCK 8

// One wave handles one 16-row tile of P (rows), loops over all 16-col tiles of Q.
// For each of its 16 rows it computes min over all 4096 Q points of ||p-q||^2
// (via d2 = |p|^2 + |q|^2 - 2 p.q, cross term from v_wmma_f32_16x16x4_f32),
// sums the 16 row-mins and atomically accumulates into accum[0].
__global__ __launch_bounds__(WAVES_PER_BLOCK * 32)
void chamfer_dir_kernel(const float* __restrict__ P,
                        const float* __restrict__ Q,
                        float* __restrict__ accum)
{
    const int lane    = threadIdx.x & 31;
    const int wave    = threadIdx.x >> 5;
    const int rowTile = blockIdx.x * WAVES_PER_BLOCK + wave;
    const int b       = blockIdx.y;
    const int i0      = rowTile * TILE;

    const float* Pb = P + (size_t)b * NPTS * 3;
    const float* Qb = Q + (size_t)b * NPTS * 3;

    const int half = lane >> 4;   // 0 = lanes 0-15 (K0,K1), 1 = lanes 16-31 (K2,K3)
    const int l15  = lane & 15;

    // ---- Load A-matrix (16x4 f32, K padded 3->4 with zero) ----
    // lanes 0-15: M=lane, (K0,K1) = (x,y); lanes 16-31: M=lane-16, (K2,K3) = (z,0)
    const float* pr = Pb + (size_t)(i0 + l15) * 3;
    const float px = pr[0], py = pr[1], pz = pr[2];
    v2f A;
    A.x = half ? pz : px;
    A.y = half ? 0.0f : py;
    const float sq_own = px * px + py * py + pz * pz;   // |p|^2 of row i0 + (lane&15)

    // C/D layout: VGPR r -> row r (lanes 0-15) / row r+8 (lanes 16-31).
    // Broadcast the matching |p|^2 per output register.
    float sq1s[8];
    #pragma unroll
    for (int r = 0; r < 8; ++r)
        sq1s[r] = __shfl(sq_own, r + (half << 3), 32);

    float minv[8];
    #pragma unroll
    for (int r = 0; r < 8; ++r) minv[r] = 3.402823466e38f;

    #pragma unroll 2
    for (int jt = 0; jt < NTILES; ++jt) {
        // ---- Load B-matrix (4x16 f32): same per-lane pattern as A ----
        const float* qr = Qb + (size_t)(jt * TILE + l15) * 3;
        const float qx = qr[0], qy = qr[1], qz = qr[2];
        v2f Bv;
        Bv.x = half ? qz : qx;
        Bv.y = half ? 0.0f : qy;
        const float sq2_own = qx * qx + qy * qy + qz * qz;  // |q|^2 of this lane's column

        v8f c = {};
        // D = A x B (+0): emits v_wmma_f32_16x16x4_f32
        c = __builtin_amdgcn_wmma_f32_16x16x4_f32(
                /*neg_a=*/false, A, /*neg_b=*/false, Bv,
                /*c_mod=*/(short)0, c, /*reuse_a=*/false, /*reuse_b=*/false);

        #pragma unroll
        for (int r = 0; r < 8; ++r) {
            const float d2 = fmaf(-2.0f, c[r], sq1s[r] + sq2_own);
            minv[r] = fminf(minv[r], d2);
        }
    }

    // Min over the 16 columns held within each 16-lane half (per output VGPR),
    // then sum the 16 per-row minima.
    float s = 0.0f;
    #pragma unroll
    for (int r = 0; r < 8; ++r) {
        float m = minv[r];
        m = fminf(m, __shfl_xor(m, 1, 32));
        m = fminf(m, __shfl_xor(m, 2, 32));
        m = fminf(m, __shfl_xor(m, 4, 32));
        m = fminf(m, __shfl_xor(m, 8, 32));
        s += m;                       // half 0: rows r, half 1: rows r+8
    }
    s += __shfl_xor(s, 16, 32);       // combine halves: sum of all 16 row-mins

    if (lane == 0) atomicAdd(accum, s);
}

__global__ void chamfer_zero_kernel(float* w)
{
    if (threadIdx.x == 0) w[0] = 0.0f;
}

__global__ void chamfer_finalize_kernel(const float* __restrict__ w,
                                        float* __restrict__ out)
{
    if (threadIdx.x == 0)
        out[0] = 100.0f * 0.5f * w[0] / ((float)NBATCH * (float)NPTS);
}

extern "C" void kernel_launch(void* const* d_in, const int* in_sizes, int n_in,
                              void* d_out, int out_size, void* d_ws, size_t ws_size,
                              hipStream_t stream)
{
    const float* a1 = (const float*)d_in[0];
    const float* a2 = (const float*)d_in[1];
    float* out = (float*)d_out;
    float* ws  = (float*)d_ws;

    chamfer_zero_kernel<<<1, 32, 0, stream>>>(ws);

    dim3 grid(NTILES / WAVES_PER_BLOCK, NBATCH);   // (32, 4)
    dim3 block(WAVES_PER_BLOCK * 32);              // 256 threads = 8 waves

    // Pass 1: for each a1 point, min over a2 points (av_dist2)
    chamfer_dir_kernel<<<grid, block, 0, stream>>>(a1, a2, ws);
    // Pass 2: for each a2 point, min over a1 points (av_dist1)
    chamfer_dir_kernel<<<grid, block, 0, stream>>>(a2, a1, ws);

    chamfer_finalize_kernel<<<1, 32, 0, stream>>>(ws, out);
}